// GCN_8160437862602
// MI455X (gfx1250) — compile-verified
//
#include <hip/hip_runtime.h>
#include <hip/hip_bf16.h>

#define N_NODES 100000
#define F_DIM   128

typedef __attribute__((ext_vector_type(2))) float v2f;
typedef __attribute__((ext_vector_type(8))) float v8f;

// ---------------------------------------------------------------- zero fill
__global__ __launch_bounds__(256) void gcn_zero(float* __restrict__ p, long long n) {
    long long i = (long long)blockIdx.x * blockDim.x + threadIdx.x;
    long long stride = (long long)gridDim.x * blockDim.x;
    for (; i < n; i += stride) p[i] = 0.0f;
}

// ---------------------------------------------------------------- degree
__global__ __launch_bounds__(256) void gcn_degree(const int* __restrict__ row,
                                                  float* __restrict__ deg, int E) {
    int e = blockIdx.x * blockDim.x + threadIdx.x;
    if (e < E) atomicAdd(&deg[row[e]], 1.0f);
}

// ---------------------------------------------------------------- norm = clip(deg,1)^-0.5
__global__ __launch_bounds__(256) void gcn_norm(const float* __restrict__ deg,
                                                float* __restrict__ norm, int n) {
    int i = blockIdx.x * blockDim.x + threadIdx.x;
    if (i < n) {
        float d = deg[i];
        d = d < 1.0f ? 1.0f : d;
        norm[i] = rsqrtf(d);
    }
}

// ---------------------------------------------------------------- x = (h*norm) @ W^T
// One wave computes one 16x16 tile of x via V_WMMA_F32_16X16X4_F32 (K=128 in 32 steps).
// A-frag lane layout (ISA 7.12.2): lane L holds A[m0+(L&15)][k + 2*(L>>4) + {0,1}]
// B-frag lane layout (symmetric):  lane L holds B[k + 2*(L>>4) + {0,1}][n0+(L&15)]
//                                         == W[n0+(L&15)][k + 2*(L>>4) + {0,1}]  (contiguous float2)
__global__ __launch_bounds__(256) void gcn_gemm(const float* __restrict__ h,
                                                const float* __restrict__ norm,
                                                const float* __restrict__ W,
                                                float* __restrict__ x) {
    const int lane = threadIdx.x & 31;
    const int wave = threadIdx.x >> 5;        // 0..7 -> column tile
    const int m0   = blockIdx.x * 16;         // row tile base (N divisible by 16)
    const int n0   = wave * 16;
    const int r    = lane & 15;
    const int koff = (lane >> 4) * 2;         // 0 or 2

    const int   mrow = m0 + r;
    const float s    = norm[mrow];            // per-source-row scaling, fused into A
    const float* __restrict__ hp = h + (size_t)mrow * F_DIM + koff;
    const float* __restrict__ wp = W + (size_t)(n0 + r) * F_DIM + koff;

    v8f acc = {};
#pragma unroll
    for (int k = 0; k < F_DIM; k += 4) {
        v2f a = *(const v2f*)(hp + k);
        a = a * s;
        v2f b = *(const v2f*)(wp + k);
        acc = __builtin_amdgcn_wmma_f32_16x16x4_f32(
            /*neg_a=*/false, a, /*neg_b=*/false, b,
            /*c_mod=*/(short)0, acc, /*reuse_a=*/false, /*reuse_b=*/false);
    }

    // D layout: VGPR i -> lanes 0-15: (M=m0+i, N=n0+lane); lanes 16-31: (M=m0+8+i, N=n0+lane-16)
    const int om_base = m0 + (lane >> 4) * 8;
    const int on      = n0 + r;
#pragma unroll
    for (int i = 0; i < 8; ++i) {
        x[(size_t)(om_base + i) * F_DIM + on] = acc[i];
    }
}

// ---------------------------------------------------------------- scatter: out[row] += norm[row]*x[col]
// One wave per edge; 32 lanes x float4 = 128 features. x/out are L2-resident (51MB each, 192MB L2).
__global__ __launch_bounds__(256) void gcn_scatter(const int* __restrict__ row,
                                                   const int* __restrict__ col,
                                                   const float* __restrict__ x,
                                                   const float* __restrict__ norm,
                                                   float* __restrict__ out, int E) {
    const int e    = blockIdx.x * 8 + (threadIdx.x >> 5);   // E divisible by 8
    const int lane = threadIdx.x & 31;
    if (e >= E) return;
    const int   rd = row[e];
    const int   cs = col[e];
    const float s  = norm[rd];
    const float4 v = ((const float4*)(x + (size_t)cs * F_DIM))[lane];
    float* op = out + (size_t)rd * F_DIM + lane * 4;
    atomicAdd(op + 0, v.x * s);
    atomicAdd(op + 1, v.y * s);
    atomicAdd(op + 2, v.z * s);
    atomicAdd(op + 3, v.w * s);
}

// ---------------------------------------------------------------- launcher
extern "C" void kernel_launch(void* const* d_in, const int* in_sizes, int n_in,
                              void* d_out, int out_size, void* d_ws, size_t ws_size,
                              hipStream_t stream) {
    const int*   row  = (const int*)  d_in[0];
    const int*   col  = (const int*)  d_in[1];
    const float* h    = (const float*)d_in[2];
    const float* W    = (const float*)d_in[3];
    float*       out  = (float*)d_out;

    const int E = in_sizes[0];                 // 1,600,000
    const int N = N_NODES;                     // 100,000

    // workspace partition: deg[N] | norm[N] | x[N*128]
    float* deg  = (float*)d_ws;
    float* norm = deg + N;
    float* x    = norm + N;

    // 1. zero accumulators (harness poisons d_out/d_ws)
    gcn_zero<<<2048, 256, 0, stream>>>(out, (long long)N * F_DIM);
    gcn_zero<<<512,  256, 0, stream>>>(deg, (long long)N);

    // 2. degrees
    gcn_degree<<<(E + 255) / 256, 256, 0, stream>>>(row, deg, E);

    // 3. norm
    gcn_norm<<<(N + 255) / 256, 256, 0, stream>>>(deg, norm, N);

    // 4. GEMM x = (h*norm) @ W^T  — fp32 WMMA, one 16x16 tile per wave, 8 waves/block
    gcn_gemm<<<N / 16, 256, 0, stream>>>(h, norm, W, x);

    // 5. scatter-add with destination norm fused
    gcn_scatter<<<E / 8, 256, 0, stream>>>(row, col, x, norm, out, E);
}